// RGCNLayer_83167746719886
// MI455X (gfx1250) — compile-verified
//
#include <hip/hip_runtime.h>
#include <stdint.h>

typedef __attribute__((ext_vector_type(2))) float v2f;
typedef __attribute__((ext_vector_type(8))) float v8f;

#define D 128
#define NREL 16
#define NBASE 8
#define LDS_STRIDE 132   // 128 + 4 floats pad -> conflict-free column reads

__device__ __forceinline__ v8f wmma4(v2f a, v2f b, v8f c) {
  // D(16x16 f32) = A(16x4 f32) * B(4x16 f32) + C
  return __builtin_amdgcn_wmma_f32_16x16x4_f32(false, a, false, b, (short)0, c,
                                               false, false);
}

// ---------- binning: count / scan / fill ----------
__global__ void rgcn_zero_kernel(int* cnt) {
  if (threadIdx.x < NREL) cnt[threadIdx.x] = 0;
}

__global__ void rgcn_count_kernel(const int* __restrict__ etype, int E,
                                  int* __restrict__ cnt) {
  __shared__ int l[NREL];
  if (threadIdx.x < NREL) l[threadIdx.x] = 0;
  __syncthreads();
  int e = blockIdx.x * blockDim.x + threadIdx.x;
  if (e < E) atomicAdd(&l[etype[e] & (NREL - 1)], 1);
  __syncthreads();
  if (threadIdx.x < NREL && l[threadIdx.x])
    atomicAdd(&cnt[threadIdx.x], l[threadIdx.x]);
}

__global__ void rgcn_scan_kernel(const int* __restrict__ cnt,
                                 int* __restrict__ relOff,
                                 int* __restrict__ blkOff,
                                 int* __restrict__ fillPos) {
  if (threadIdx.x == 0) {
    int ro = 0, bo = 0;
    for (int r = 0; r < NREL; ++r) {
      relOff[r] = ro;
      blkOff[r] = bo;
      fillPos[r] = ro;
      ro += cnt[r];
      bo += (cnt[r] + 15) >> 4;
    }
    relOff[NREL] = ro;
    blkOff[NREL] = bo;
  }
}

__global__ void rgcn_fill_kernel(const int* __restrict__ etype, int E,
                                 int* __restrict__ fillPos,
                                 int* __restrict__ perm) {
  int e = blockIdx.x * blockDim.x + threadIdx.x;
  if (e < E) {
    int pos = atomicAdd(&fillPos[etype[e] & (NREL - 1)], 1);
    perm[pos] = e;
  }
}

// ---------- W[r] = sum_b coef[r,b] * bases[b] ----------
__global__ void rgcn_wcomp_kernel(const float* __restrict__ coef,
                                  const float* __restrict__ bases,
                                  float* __restrict__ W) {
  int idx = blockIdx.x * blockDim.x + threadIdx.x;
  if (idx < NREL * D * D) {
    int r = idx >> 14;          // /(128*128)
    int io = idx & (D * D - 1);
    float acc = 0.f;
#pragma unroll
    for (int b = 0; b < NBASE; ++b)
      acc += coef[r * NBASE + b] * bases[(size_t)b * D * D + io];
    W[idx] = acc;
  }
}

// ---------- self loop: out = x @ w_self^T (initializes out) ----------
__global__ __launch_bounds__(256) void rgcn_self_kernel(
    const float* __restrict__ x, const float* __restrict__ wself,
    float* __restrict__ out, int N, int blocksPerWG) {
  __shared__ __align__(16) float lds_x[16 * LDS_STRIDE];
  const int tid = threadIdx.x;
  const int lane = tid & 31;
  const int wave = tid >> 5;   // column tile 0..7
  const int half = lane >> 4;  // 0/1 -> K pair select
  const int l15 = lane & 15;

  int nblk = (N + 15) >> 4;
  int blk0 = blockIdx.x * blocksPerWG;
  if (blk0 >= nblk) return;
  int blkEnd = min(blk0 + blocksPerWG, nblk);

  // B[k][n] = w_self[n][k]   (register-resident for entire kernel)
  v2f bfrag[32];
  {
    int n = wave * 16 + l15;
#pragma unroll
    for (int kc = 0; kc < 32; ++kc) {
      int kk = kc * 4 + half * 2;
      float2 w2 = *(const float2*)(wself + (size_t)n * D + kk);
      v2f b;
      b.x = w2.x;
      b.y = w2.y;
      bfrag[kc] = b;
    }
  }

  for (int blk = blk0; blk < blkEnd; ++blk) {
    int n0 = blk * 16;
    // stage 16 contiguous rows of x into LDS (coalesced float4)
    int j = tid >> 4;
    int cpos = (tid & 15) * 8;
    int srow = min(n0 + j, N - 1);
    const float4* sp = (const float4*)(x + (size_t)srow * D + cpos);
    float4 v0 = sp[0];
    float4 v1 = sp[1];
    float* dp = &lds_x[j * LDS_STRIDE + cpos];
    *(float4*)dp = v0;
    *(float4*)(dp + 4) = v1;
    __syncthreads();

    v8f acc = {0.f, 0.f, 0.f, 0.f, 0.f, 0.f, 0.f, 0.f};
#pragma unroll
    for (int kc = 0; kc < 32; ++kc) {
      int kk = kc * 4 + half * 2;
      const float* ap = &lds_x[l15 * LDS_STRIDE + kk];
      v2f a;
      a.x = ap[0];
      a.y = ap[1];
      acc = wmma4(a, bfrag[kc], acc);
    }

    int n = wave * 16 + l15;
#pragma unroll
    for (int v = 0; v < 8; ++v) {
      int m = n0 + v + half * 8;
      if (m < N) out[(size_t)m * D + n] = acc[v];
    }
    __syncthreads();
  }
}

// ---------- grouped gather-GEMM-scatter over relation-binned edges ----------
__global__ __launch_bounds__(256) void rgcn_edge_kernel(
    const float* __restrict__ x, const int* __restrict__ eidx,  // [2*E]
    const float* __restrict__ W, const int* __restrict__ relOff,
    const int* __restrict__ blkOff, const int* __restrict__ perm,
    float* __restrict__ out, int E, int blocksPerWG) {
  __shared__ __align__(16) float lds_x[16 * LDS_STRIDE];
  __shared__ int ldsDest[16];

  const int tid = threadIdx.x;
  const int lane = tid & 31;
  const int wave = tid >> 5;
  const int half = lane >> 4;
  const int l15 = lane & 15;

  const int totalBlk = blkOff[NREL];
  int blk0 = blockIdx.x * blocksPerWG;
  if (blk0 >= totalBlk) return;
  int blkEnd = min(blk0 + blocksPerWG, totalBlk);

  v2f bfrag[32];
  int curR = -1;

  for (int blk = blk0; blk < blkEnd; ++blk) {
    // relation of this block (blocks sorted by relation; skips empty rels)
    int r = 0;
    while (r < NREL - 1 && blk >= blkOff[r + 1]) ++r;

    if (r != curR) {  // (re)load relation matrix tile into registers
      curR = r;
      const float* Wr = W + (size_t)r * (D * D);
      int n = wave * 16 + l15;
#pragma unroll
      for (int kc = 0; kc < 32; ++kc) {
        int kk = kc * 4 + half * 2;
        v2f b;
        b.x = Wr[kk * D + n];
        b.y = Wr[(kk + 1) * D + n];
        bfrag[kc] = b;
      }
    }

    int base = relOff[r] + (blk - blkOff[r]) * 16;
    int gend = relOff[r + 1];

    // gather 16 source rows into LDS
    int j = tid >> 4;
    int cpos = (tid & 15) * 8;
    int ec = perm[min(base + j, gend - 1)];
    int src = eidx[E + ec];  // edge_index[1] = source node
    const float4* sp = (const float4*)(x + (size_t)src * D + cpos);
    float4 v0 = sp[0];
    float4 v1 = sp[1];
    float* dp = &lds_x[j * LDS_STRIDE + cpos];
    *(float4*)dp = v0;
    *(float4*)(dp + 4) = v1;
    if (tid < 16) {
      int ie = base + tid;
      ldsDest[tid] = (ie < gend) ? eidx[perm[ie]] : -1;  // edge_index[0] = dest
    }
    __syncthreads();

    v8f acc = {0.f, 0.f, 0.f, 0.f, 0.f, 0.f, 0.f, 0.f};
#pragma unroll
    for (int kc = 0; kc < 32; ++kc) {
      int kk = kc * 4 + half * 2;
      const float* ap = &lds_x[l15 * LDS_STRIDE + kk];
      v2f a;
      a.x = ap[0];
      a.y = ap[1];
      acc = wmma4(a, bfrag[kc], acc);
    }

    // scatter-add message tile to destination rows
    int n = wave * 16 + l15;
#pragma unroll
    for (int v = 0; v < 8; ++v) {
      int dest = ldsDest[v + half * 8];
      if (dest >= 0) atomicAdd(&out[(size_t)dest * D + n], acc[v]);
    }
    __syncthreads();
  }
}

extern "C" void kernel_launch(void* const* d_in, const int* in_sizes, int n_in,
                              void* d_out, int out_size, void* d_ws,
                              size_t ws_size, hipStream_t stream) {
  const float* x = (const float*)d_in[0];
  const int* eidx = (const int*)d_in[1];   // [2, E]
  const int* etype = (const int*)d_in[2];  // [E]
  const float* bases = (const float*)d_in[3];
  const float* coef = (const float*)d_in[4];
  const float* wself = (const float*)d_in[5];
  float* out = (float*)d_out;

  int N = in_sizes[0] / D;
  int E = in_sizes[2];

  // workspace layout: W[16*128*128] f32 | cnt[16] relOff[17] blkOff[17]
  //                   fillPos[16] perm[E]
  float* W = (float*)d_ws;
  int* ib = (int*)((char*)d_ws + (size_t)NREL * D * D * sizeof(float));
  int* cnt = ib;
  int* relOff = ib + 16;
  int* blkOff = ib + 33;
  int* fillPos = ib + 50;
  int* perm = ib + 66;

  dim3 b256(256);
  rgcn_zero_kernel<<<1, 64, 0, stream>>>(cnt);
  rgcn_count_kernel<<<(E + 255) / 256, b256, 0, stream>>>(etype, E, cnt);
  rgcn_scan_kernel<<<1, 32, 0, stream>>>(cnt, relOff, blkOff, fillPos);
  rgcn_fill_kernel<<<(E + 255) / 256, b256, 0, stream>>>(etype, E, fillPos,
                                                         perm);
  rgcn_wcomp_kernel<<<(NREL * D * D + 255) / 256, b256, 0, stream>>>(coef,
                                                                     bases, W);

  int nblkSelf = (N + 15) / 16;
  int selfPerWG = 8;
  rgcn_self_kernel<<<(nblkSelf + selfPerWG - 1) / selfPerWG, b256, 0, stream>>>(
      x, wself, out, N, selfPerWG);

  int tbMax = (E + 15) / 16 + NREL;  // upper bound on total blocks
  int edgePerWG = 16;
  rgcn_edge_kernel<<<(tbMax + edgePerWG - 1) / edgePerWG, b256, 0, stream>>>(
      x, eidx, W, relOff, blkOff, perm, out, E, edgePerWG);
}